// Spatial_Temporal_Att_36558761623652
// MI455X (gfx1250) — compile-verified
//
#include <hip/hip_runtime.h>
#include <math.h>

// Spatial_Temporal_Att: pooled = [mean_c(x), max_c(x)]; out = sigmoid(conv3d(pooled, w) + b)
// x: [8, 64, 32, 64, 64] f32; w: [1,2,3,3,3]; b: [1]; out: [8,1,32,64,64]

typedef float v2f __attribute__((ext_vector_type(2)));
typedef float v8f __attribute__((ext_vector_type(8)));

#define S_SPATIAL (32 * 64 * 64)      /* 131072 positions per batch   */
#define C_IN      64
#define NPOS      (8 * S_SPATIAL)     /* 1,048,576 output positions   */

// -------- Pass 1: channel mean (WMMA f32 16x16x4, A = ones) + channel max ----
// One wave handles a tile of 16 consecutive positions x 64 channels.
//   B operand per 4-channel chunk: lane L -> position = tile + (L & 15),
//   lanes 0-15 supply channels {c, c+1}, lanes 16-31 supply {c+2, c+3}
//   (16x4 A-layout mirrored: VGPR0 holds K=0/K=2, VGPR1 holds K=1/K=3).
// After 16 accumulating WMMAs, D VGPR0 of lane L = sum over all 64 channels
// at position (L & 15) -> lane-aligned, coalesced 64B stores.
__global__ __launch_bounds__(256) void pool_meanmax_kernel(
    const float* __restrict__ x, float* __restrict__ pooled) {
  const int lane = threadIdx.x & 31;
  const int wave = threadIdx.x >> 5;
  const int q0   = (blockIdx.x * 8 + wave) * 16;      // 16-aligned; never crosses b
  const int b    = q0 / S_SPATIAL;
  const int p    = (q0 % S_SPATIAL) + (lane & 15);
  const int g    = lane >> 4;                          // half-wave selector
  const float* xb = x + (size_t)b * C_IN * S_SPATIAL + p;

  v8f acc = {0.f, 0.f, 0.f, 0.f, 0.f, 0.f, 0.f, 0.f};
  const v2f aones = {1.f, 1.f};
  float mx = -INFINITY;

#pragma unroll
  for (int ck = 0; ck < 16; ++ck) {
    const int c0 = ck * 4 + g * 2;
    v2f bv;
    bv.x = xb[(size_t)(c0)     * S_SPATIAL];
    bv.y = xb[(size_t)(c0 + 1) * S_SPATIAL];
    mx = fmaxf(mx, fmaxf(bv.x, bv.y));
    // (neg_a, A, neg_b, B, c_mod, C, reuse_a, reuse_b)
    acc = __builtin_amdgcn_wmma_f32_16x16x4_f32(false, aones, false, bv,
                                                (short)0, acc, false, false);
  }
  // half-wave g=0 saw channels {0,1 mod 4}, g=1 saw {2,3 mod 4}: combine maxima
  mx = fmaxf(mx, __shfl_xor(mx, 16, 32));

  const float mean = acc[0] * (1.0f / 64.0f);
  float* pb = pooled + (size_t)b * 2 * S_SPATIAL;
  if (g == 0) pb[p] = mean;             // pooled[b][0][p]  (avg)
  else        pb[S_SPATIAL + p] = mx;   // pooled[b][1][p]  (max)
}

// -------- Pass 2: 3x3x3 conv (2->1 ch, pad 1) + bias + sigmoid ---------------
__global__ __launch_bounds__(256) void conv3d_sigmoid_kernel(
    const float* __restrict__ pooled, const float* __restrict__ wgt,
    const float* __restrict__ bias, float* __restrict__ out) {
  __shared__ float sw[54];
  __shared__ float sb;
  if (threadIdx.x < 54) sw[threadIdx.x] = wgt[threadIdx.x];
  if (threadIdx.x == 54) sb = bias[0];
  __syncthreads();

  const int q = blockIdx.x * 256 + threadIdx.x;
  const int b = q / S_SPATIAL;
  int r = q % S_SPATIAL;
  const int d  = r >> 12;          // / (64*64)
  r &= 4095;
  const int h  = r >> 6;
  const int wp = r & 63;

  const float* pb = pooled + (size_t)b * 2 * S_SPATIAL;
  float acc = sb;
#pragma unroll
  for (int ic = 0; ic < 2; ++ic) {
#pragma unroll
    for (int kd = 0; kd < 3; ++kd) {
      const int zd = d + kd - 1;
      if (zd < 0 || zd >= 32) continue;
#pragma unroll
      for (int kh = 0; kh < 3; ++kh) {
        const int zh = h + kh - 1;
        if (zh < 0 || zh >= 64) continue;
#pragma unroll
        for (int kw = 0; kw < 3; ++kw) {
          const int zw = wp + kw - 1;
          if (zw < 0 || zw >= 64) continue;
          acc += sw[((ic * 3 + kd) * 3 + kh) * 3 + kw] *
                 pb[(size_t)ic * S_SPATIAL + ((zd << 6) + zh) * 64 + zw];
        }
      }
    }
  }
  out[q] = 1.0f / (1.0f + expf(-acc));
}

extern "C" void kernel_launch(void* const* d_in, const int* in_sizes, int n_in,
                              void* d_out, int out_size, void* d_ws, size_t ws_size,
                              hipStream_t stream) {
  const float* x = (const float*)d_in[0];   // [8,64,32,64,64]
  const float* w = (const float*)d_in[1];   // [1,2,3,3,3]
  const float* b = (const float*)d_in[2];   // [1]
  float* out    = (float*)d_out;            // [8,1,32,64,64]
  float* pooled = (float*)d_ws;             // [8,2,32,64,64] = 8 MB scratch

  // Pass 1: 8192 blocks x 8 waves x 16 positions = all 1M positions
  pool_meanmax_kernel<<<NPOS / 128, 256, 0, stream>>>(x, pooled);
  // Pass 2: one thread per output position
  conv3d_sigmoid_kernel<<<NPOS / 256, 256, 0, stream>>>(pooled, w, b, out);
}